// GraphNN_48825188221302
// MI455X (gfx1250) — compile-verified
//
#include <hip/hip_runtime.h>
#include <math.h>

// ---------------- problem constants (match reference) ----------------
#define NN      100000
#define TT      4
#define EE      250000
#define DD      64
#define HH      4
#define CC      64
#define F_IN    256
#define LL      2
#define ETOT    (EE + NN)          // edges + self loops
#define HC      (HH * CC)          // 256
#define NEG_SLOPE 0.2f

typedef float v2f __attribute__((ext_vector_type(2)));
typedef float v8f __attribute__((ext_vector_type(8)));

__device__ __forceinline__ float gelu_exact(float v) {
    return 0.5f * v * (1.0f + erff(v * 0.70710678118654752440f));
}
__device__ __forceinline__ float lrelu(float v) {
    return v > 0.0f ? v : NEG_SLOPE * v;
}
// float atomic max via sign-split bit trick (segmax initialized to -inf)
__device__ __forceinline__ void atomicMaxF(float* addr, float v) {
    if (v >= 0.0f) atomicMax((int*)addr, __float_as_int(v));
    else           atomicMin((unsigned int*)addr, __float_as_uint(v));
}

// ---------------------------------------------------------------------
// Fold attention vectors into weights:
//   wsv[b, d, h] = sum_c W_src[b, d, h*C+c] * a_src[b, h, c]   (b = l*T+t)
// so s_src[n,h] = x[n,:] . wsv[b,:,h]  (avoids materializing xd entirely
// and avoids re-reading xs for scores).
// ---------------------------------------------------------------------
__global__ __launch_bounds__(256)
void fold_kernel(const float* __restrict__ Wsrc, const float* __restrict__ Wdst,
                 const float* __restrict__ asrc, const float* __restrict__ adst,
                 float* __restrict__ wsv, float* __restrict__ wdv) {
    const int b   = blockIdx.x;            // l*T + t  (0..7)
    const int tid = threadIdx.x;           // 256 = 64 d * 4 h
    const int d = tid >> 2, h = tid & 3;
    const float* Ws = Wsrc + (size_t)b * DD * HC;
    const float* Wd = Wdst + (size_t)b * DD * HC;
    const float* As = asrc + (size_t)b * HH * CC;
    const float* Ad = adst + (size_t)b * HH * CC;
    float s = 0.0f, t = 0.0f;
    #pragma unroll 4
    for (int c = 0; c < CC; ++c) {
        s += Ws[d * HC + h * CC + c] * As[h * CC + c];
        t += Wd[d * HC + h * CC + c] * Ad[h * CC + c];
    }
    wsv[b * (DD * HH) + d * HH + h] = s;
    wdv[b * (DD * HH) + d * HH + h] = t;
}

// ---------------------------------------------------------------------
// f32 WMMA GEMM:  C[M,Nn] = A[M,K] @ B[K,Nn]  (+ optional bias per col)
// Block = 128 threads = 4 waves; block tile = 16 rows x 64 cols
// (one 16x16 WMMA tile per wave); grid = (M/16, Nn/64).
// A tile staged in LDS (16*K floats, contiguous rows).
// V_WMMA_F32_16X16X4_F32 operand striping (wave32):
//   A: lane&15 = row, a[v] = A[row, k0 + 2*(lane>>4) + v]
//   B: lane&15 = col, b[v] = B[k0 + 2*(lane>>4) + v, col]
//   C: vgpr r -> row r + 8*(lane>>4), col = lane&15
// ---------------------------------------------------------------------
__global__ __launch_bounds__(128)
void gemm_wmma_f32(const float* __restrict__ A, const float* __restrict__ B,
                   const float* __restrict__ bias, float* __restrict__ Cout,
                   int K, int Nn) {
    __shared__ float At[16 * 256];         // max K = 256 -> 16 KB
    const int m0   = blockIdx.x * 16;
    const int tid  = threadIdx.x;
    const int wave = tid >> 5;
    const int lane = tid & 31;

    // cooperative stage of contiguous 16*K A-tile
    const float4* src4 = (const float4*)(A + (size_t)m0 * K);
    float4*       dst4 = (float4*)At;
    const int n4 = 4 * K;                  // (16*K)/4
    for (int i = tid; i < n4; i += 128) dst4[i] = src4[i];
    __syncthreads();

    const int lo   = lane & 15;
    const int hi   = lane >> 4;
    const int col0 = blockIdx.y * 64 + wave * 16;

    v8f c = {};
    for (int k0 = 0; k0 < K; k0 += 4) {
        const int kk = k0 + 2 * hi;
        v2f a, b;
        a.x = At[lo * K + kk];
        a.y = At[lo * K + kk + 1];
        b.x = B[(size_t)kk       * Nn + col0 + lo];
        b.y = B[(size_t)(kk + 1) * Nn + col0 + lo];
        c = __builtin_amdgcn_wmma_f32_16x16x4_f32(
                false, a, false, b, (short)0, c, false, false);
    }
    const float bb = bias ? bias[col0 + lo] : 0.0f;
    #pragma unroll
    for (int r = 0; r < 8; ++r) {
        const int row = m0 + r + hi * 8;
        Cout[(size_t)row * Nn + col0 + lo] = c[r] + bb;
    }
}

// ---------------------------------------------------------------------
// x = GELU(LayerNorm(xf_raw)) + L2Normalize(emb)     (wave per node)
// ---------------------------------------------------------------------
__global__ __launch_bounds__(256)
void posttransform_kernel(const float* __restrict__ xf, const float* __restrict__ emb,
                          const float* __restrict__ g, const float* __restrict__ be,
                          float* __restrict__ xout) {
    const int n    = (blockIdx.x * blockDim.x + threadIdx.x) >> 5;
    const int lane = threadIdx.x & 31;
    if (n >= NN) return;
    float2 v = ((const float2*)(xf + (size_t)n * DD))[lane];
    float s1 = v.x + v.y;
    float s2 = v.x * v.x + v.y * v.y;
    #pragma unroll
    for (int o = 16; o; o >>= 1) { s1 += __shfl_xor(s1, o); s2 += __shfl_xor(s2, o); }
    const float mu   = s1 * (1.0f / 64.0f);
    const float var  = s2 * (1.0f / 64.0f) - mu * mu;
    const float rstd = rsqrtf(var + 1e-5f);
    const int c = lane * 2;
    float a0 = gelu_exact((v.x - mu) * rstd * g[c]     + be[c]);
    float a1 = gelu_exact((v.y - mu) * rstd * g[c + 1] + be[c + 1]);
    float2 e = ((const float2*)(emb + (size_t)n * DD))[lane];
    float ss = e.x * e.x + e.y * e.y;
    #pragma unroll
    for (int o = 16; o; o >>= 1) ss += __shfl_xor(ss, o);
    const float inv = 1.0f / fmaxf(sqrtf(ss), 1e-12f);
    float2 out; out.x = a0 + e.x * inv; out.y = a1 + e.y * inv;
    ((float2*)(xout + (size_t)n * DD))[lane] = out;
}

// ---------------------------------------------------------------------
// Per-node attention scores via folded weights (wave per node):
//   ssrc[n,h] = x[n,:] . wsv[:,h] ; sdst[n,h] = x[n,:] . wdv[:,h]
// ---------------------------------------------------------------------
__global__ __launch_bounds__(256)
void score_kernel(const float* __restrict__ x, const float* __restrict__ wsv,
                  const float* __restrict__ wdv, float* __restrict__ ssrc,
                  float* __restrict__ sdst) {
    const int n    = (blockIdx.x * blockDim.x + threadIdx.x) >> 5;
    const int lane = threadIdx.x & 31;
    if (n >= NN) return;
    float2 v = ((const float2*)(x + (size_t)n * DD))[lane];
    const int d0 = lane * 2;
    float s[HH], t[HH];
    #pragma unroll
    for (int h = 0; h < HH; ++h) {
        s[h] = v.x * wsv[d0 * HH + h] + v.y * wsv[(d0 + 1) * HH + h];
        t[h] = v.x * wdv[d0 * HH + h] + v.y * wdv[(d0 + 1) * HH + h];
    }
    #pragma unroll
    for (int h = 0; h < HH; ++h)
        #pragma unroll
        for (int o = 16; o; o >>= 1) {
            s[h] += __shfl_xor(s[h], o);
            t[h] += __shfl_xor(t[h], o);
        }
    if (lane == 0) {
        ((float4*)ssrc)[n] = make_float4(s[0], s[1], s[2], s[3]);
        ((float4*)sdst)[n] = make_float4(t[0], t[1], t[2], t[3]);
    }
}

__global__ __launch_bounds__(256)
void init_seg_kernel(float* __restrict__ segmax, float* __restrict__ segsum) {
    const int i = blockIdx.x * blockDim.x + threadIdx.x;
    if (i < NN * HH) { segmax[i] = -INFINITY; segsum[i] = 0.0f; }
}

// pass 1: segment max of leaky-relu logits over dst
__global__ __launch_bounds__(256)
void edge_max_kernel(const int* __restrict__ ei, const float* __restrict__ ssrc,
                     const float* __restrict__ sdst, float* __restrict__ segmax) {
    const int e = blockIdx.x * blockDim.x + threadIdx.x;
    if (e >= ETOT) return;
    int src, dst;
    if (e < EE) { src = ei[e]; dst = ei[EE + e]; }
    else        { src = dst = e - EE; }
    const float4 a = ((const float4*)ssrc)[src];
    const float4 b = ((const float4*)sdst)[dst];
    atomicMaxF(&segmax[dst * HH + 0], lrelu(a.x + b.x));
    atomicMaxF(&segmax[dst * HH + 1], lrelu(a.y + b.y));
    atomicMaxF(&segmax[dst * HH + 2], lrelu(a.z + b.z));
    atomicMaxF(&segmax[dst * HH + 3], lrelu(a.w + b.w));
}

// pass 2: segment sum of exp(logit - max)
__global__ __launch_bounds__(256)
void edge_sum_kernel(const int* __restrict__ ei, const float* __restrict__ ssrc,
                     const float* __restrict__ sdst, const float* __restrict__ segmax,
                     float* __restrict__ segsum) {
    const int e = blockIdx.x * blockDim.x + threadIdx.x;
    if (e >= ETOT) return;
    int src, dst;
    if (e < EE) { src = ei[e]; dst = ei[EE + e]; }
    else        { src = dst = e - EE; }
    const float4 a = ((const float4*)ssrc)[src];
    const float4 b = ((const float4*)sdst)[dst];
    const float4 m = ((const float4*)segmax)[dst];
    atomicAdd(&segsum[dst * HH + 0], expf(lrelu(a.x + b.x) - m.x));
    atomicAdd(&segsum[dst * HH + 1], expf(lrelu(a.y + b.y) - m.y));
    atomicAdd(&segsum[dst * HH + 2], expf(lrelu(a.z + b.z) - m.z));
    atomicAdd(&segsum[dst * HH + 3], expf(lrelu(a.w + b.w) - m.w));
}

// pass 3: acc[dst,c] += (1/H) * sum_h alpha[e,h] * xs[src, h*C+c]
// 64 threads per edge (head mean folded in -> 64 atomics/edge, not 256)
__global__ __launch_bounds__(256)
void edge_scatter_kernel(const int* __restrict__ ei, const float* __restrict__ ssrc,
                         const float* __restrict__ sdst, const float* __restrict__ segmax,
                         const float* __restrict__ segsum, const float* __restrict__ xs,
                         float* __restrict__ acc) {
    __shared__ float alpha_sh[4][HH];
    const int g = threadIdx.x >> 6;        // edge slot in block (0..3)
    const int c = threadIdx.x & 63;        // channel
    const int e = blockIdx.x * 4 + g;
    const bool valid = (e < ETOT);
    int src = 0, dst = 0;
    if (valid) {
        if (e < EE) { src = ei[e]; dst = ei[EE + e]; }
        else        { src = dst = e - EE; }
        if (c < HH) {
            const float lv = lrelu(ssrc[src * HH + c] + sdst[dst * HH + c]);
            alpha_sh[g][c] = expf(lv - segmax[dst * HH + c]) / segsum[dst * HH + c];
        }
    }
    __syncthreads();
    if (!valid) return;
    const float* xr = xs + (size_t)src * HC;
    const float v = alpha_sh[g][0] * xr[c]
                  + alpha_sh[g][1] * xr[CC + c]
                  + alpha_sh[g][2] * xr[2 * CC + c]
                  + alpha_sh[g][3] * xr[3 * CC + c];
    atomicAdd(&acc[(size_t)dst * CC + c], v * 0.25f);   // mean over H=4
}

// x_new = GELU( (acc + sum_t bias[l,t,:]) / T )
__global__ __launch_bounds__(256)
void finalize_kernel(const float* __restrict__ acc, const float* __restrict__ bias_l,
                     float* __restrict__ xout) {
    const int i = blockIdx.x * blockDim.x + threadIdx.x;
    if (i >= NN * CC) return;
    const int c = i & 63;
    const float b = bias_l[c] + bias_l[CC + c] + bias_l[2 * CC + c] + bias_l[3 * CC + c];
    xout[i] = gelu_exact((acc[i] + b) * 0.25f);          // /T, T=4
}

// ---------------------------------------------------------------------
extern "C" void kernel_launch(void* const* d_in, const int* in_sizes, int n_in,
                              void* d_out, int out_size, void* d_ws, size_t ws_size,
                              hipStream_t stream) {
    const float* feat   = (const float*)d_in[0];
    const float* emb    = (const float*)d_in[1];
    const float* lin_w  = (const float*)d_in[2];
    const float* lin_b  = (const float*)d_in[3];
    const float* ln_g   = (const float*)d_in[4];
    const float* ln_b   = (const float*)d_in[5];
    const float* W_src  = (const float*)d_in[6];
    const float* W_dst  = (const float*)d_in[7];
    const float* a_src  = (const float*)d_in[8];
    const float* a_dst  = (const float*)d_in[9];
    const float* c_bias = (const float*)d_in[10];
    const int*   eidx   = (const int*)d_in[11];

    // workspace layout (256B aligned)
    char* ws = (char*)d_ws;
    size_t off = 0;
    auto carve = [&](size_t bytes) -> void* {
        void* p = ws + off;
        off = (off + bytes + 255) & ~(size_t)255;
        return p;
    };
    float* xbuf   = (float*)carve((size_t)NN * DD * sizeof(float));   // node features
    float* accbuf = (float*)carve((size_t)NN * CC * sizeof(float));   // conv acc / xf_raw
    float* xsbuf  = (float*)carve((size_t)NN * HC * sizeof(float));   // src projection
    float* ssrc   = (float*)carve((size_t)NN * HH * sizeof(float));
    float* sdst   = (float*)carve((size_t)NN * HH * sizeof(float));
    float* segmax = (float*)carve((size_t)NN * HH * sizeof(float));
    float* segsum = (float*)carve((size_t)NN * HH * sizeof(float));
    float* wsv    = (float*)carve((size_t)LL * TT * DD * HH * sizeof(float));
    float* wdv    = (float*)carve((size_t)LL * TT * DD * HH * sizeof(float));

    // 1) fold attention vectors into [D,H] score weights (all l,t at once)
    fold_kernel<<<LL * TT, 256, 0, stream>>>(W_src, W_dst, a_src, a_dst, wsv, wdv);

    // 2) feature GEMM: xf_raw = feat @ lin_w + lin_b   (WMMA, M=100000,K=256,Nn=64)
    gemm_wmma_f32<<<dim3(NN / 16, 1), 128, 0, stream>>>(
        feat, lin_w, lin_b, accbuf, F_IN, DD);

    // 3) x = GELU(LN(xf_raw)) + l2norm(emb)
    {
        const int threads = 256, waves_total = NN;
        const int blocks = (waves_total * 32 + threads - 1) / threads;
        posttransform_kernel<<<blocks, threads, 0, stream>>>(
            accbuf, emb, ln_g, ln_b, xbuf);
    }

    const int eb  = (ETOT + 255) / 256;
    const int nb4 = (NN * HH + 255) / 256;
    const int nbC = (NN * CC + 255) / 256;
    const int scoreb = (NN * 32 + 255) / 256;

    for (int l = 0; l < LL; ++l) {
        hipMemsetAsync(accbuf, 0, (size_t)NN * CC * sizeof(float), stream);
        for (int t = 0; t < TT; ++t) {
            const int b = l * TT + t;
            const float* Wb  = W_src + (size_t)b * DD * HC;
            const int*   eit = eidx + (size_t)t * 2 * EE;

            // source projection xs = x @ W_src[l,t]  (WMMA, K=64, Nn=256)
            gemm_wmma_f32<<<dim3(NN / 16, HC / 64), 128, 0, stream>>>(
                xbuf, Wb, nullptr, xsbuf, DD, HC);

            // per-node scores via folded weights
            score_kernel<<<scoreb, 256, 0, stream>>>(
                xbuf, wsv + b * DD * HH, wdv + b * DD * HH, ssrc, sdst);

            init_seg_kernel<<<nb4, 256, 0, stream>>>(segmax, segsum);
            edge_max_kernel<<<eb, 256, 0, stream>>>(eit, ssrc, sdst, segmax);
            edge_sum_kernel<<<eb, 256, 0, stream>>>(eit, ssrc, sdst, segmax, segsum);
            edge_scatter_kernel<<<(ETOT + 3) / 4, 256, 0, stream>>>(
                eit, ssrc, sdst, segmax, segsum, xsbuf, accbuf);
        }
        float* xout = (l == LL - 1) ? (float*)d_out : xbuf;
        finalize_kernel<<<nbC, 256, 0, stream>>>(
            accbuf, c_bias + (size_t)l * TT * CC, xout);
    }
}